// MusicGNN_17042430231417
// MI455X (gfx1250) — compile-verified
//
#include <hip/hip_runtime.h>

typedef __attribute__((ext_vector_type(2))) float v2f;
typedef __attribute__((ext_vector_type(8))) float v8f;

// ---------------------------------------------------------------------------
// CDNA5 async global->LDS copy (ASYNCcnt-tracked, bypasses VGPRs).
// GV mode: global_load_async_to_lds_b128 vdst(LDS byte addr), vaddr(64b), off
// Generic pointers to __shared__ carry the LDS byte offset in addr[31:0].
// ---------------------------------------------------------------------------
__device__ __forceinline__ void async_b128(unsigned lds_byte, const float* g) {
    asm volatile("global_load_async_to_lds_b128 %0, %1, off"
                 :: "v"(lds_byte), "v"(g) : "memory");
}
__device__ __forceinline__ void wait_async0() {
    asm volatile("s_wait_asynccnt 0x0" ::: "memory");
}
__device__ __forceinline__ unsigned lds_addr(const void* p) {
    return (unsigned)(unsigned long long)p;
}

// ---------------------------------------------------------------------------
// GCN normalization helpers
// ---------------------------------------------------------------------------
__global__ void k_fill1(float* __restrict__ deg, int n) {
    int i = blockIdx.x * blockDim.x + threadIdx.x;
    if (i < n) deg[i] = 1.0f;  // self-loop contributes 1 to every degree
}

__global__ void k_degree(const long long* __restrict__ ei, float* __restrict__ deg,
                         long long E) {
    long long e = (long long)blockIdx.x * blockDim.x + threadIdx.x;
    if (e < E) {
        int c = (int)ei[E + e];  // col = target node
        atomicAdd(&deg[c], 1.0f);
    }
}

__global__ void k_rsqrt(float* __restrict__ deg, int n) {
    int i = blockIdx.x * blockDim.x + threadIdx.x;
    if (i < n) deg[i] = rsqrtf(deg[i]);  // deg >= 1 always
}

// ---------------------------------------------------------------------------
// f32 WMMA GEMM: C[M x NOUT] = A[M x K] @ W[K x NOUT]
// block = 128 threads (4 waves), 64 rows per block:
//   RT=4 row tiles x CT=NOUT/16 col tiles, RPW=RT*CT/4 row tiles per wave.
// A tile (64 x K) and full W (K x NOUT) staged via async global->LDS b128.
// WMMA f32 16x16x4 operand layout per CDNA5 ISA 7.12.2:
//   A: lane m = lane&15; VGPR0/1 hold K = 2*hi, 2*hi+1   (hi = lane>>4)
//   B: lane n = lane&15; VGPR0/1 hold K = 2*hi, 2*hi+1
//   D: VGPR r -> row r + 8*hi, col = lane&15
// B fragment is loaded once per k-step and reused across RPW row tiles.
// ---------------------------------------------------------------------------
template <int K, int NOUT>
__global__ void __launch_bounds__(128)
k_gemm_wmma(const float* __restrict__ A, const float* __restrict__ W,
            float* __restrict__ C, int M) {
    constexpr int RT  = 4;
    constexpr int CT  = NOUT / 16;
    constexpr int RPW = RT * CT / 4;
    __shared__ __align__(16) float sW[K * NOUT];
    __shared__ __align__(16) float sA[RT * 16 * K];
    const int tid  = threadIdx.x;
    const int wave = tid >> 5;
    const int lane = tid & 31;
    const int rbase = blockIdx.x * (RT * 16);

    // stage W (contiguous K*NOUT floats) via async copy
    {
        const unsigned lw = lds_addr(sW);
        for (int c = tid; c < K * NOUT / 4; c += 128)
            async_b128(lw + c * 16, W + c * 4);
    }
    // stage A tile: rows rbase..rbase+63 are contiguous RT*16*K floats
    {
        const unsigned la = lds_addr(sA);
        const long long lim  = (long long)M * K - 4;
        const long long base = (long long)rbase * K;
        for (int c = tid; c < RT * 16 * K / 4; c += 128) {
            long long gi = base + (long long)c * 4;
            if (gi > lim) gi = lim;  // clamp tail block (rows never stored)
            async_b128(la + c * 16, A + gi);
        }
    }
    wait_async0();
    __syncthreads();

    const int ctile  = wave % CT;
    const int rwbase = (wave / CT) * RPW;
    const int hi  = lane >> 4;
    const int l15 = lane & 15;
    const int cb  = ctile * 16;

    v8f acc[RPW] = {};
#pragma unroll
    for (int k0 = 0; k0 < K; k0 += 4) {
        const int ka = k0 + 2 * hi;
        v2f b;
        b.x = sW[ka * NOUT + cb + l15];
        b.y = sW[(ka + 1) * NOUT + cb + l15];
#pragma unroll
        for (int j = 0; j < RPW; ++j) {
            const int rt = rwbase + j;
            v2f a;
            a.x = sA[(rt * 16 + l15) * K + ka];
            a.y = sA[(rt * 16 + l15) * K + ka + 1];
            acc[j] = __builtin_amdgcn_wmma_f32_16x16x4_f32(false, a, false, b,
                                                           (short)0, acc[j],
                                                           false, false);
        }
    }

#pragma unroll
    for (int j = 0; j < RPW; ++j) {
        const int rt = rwbase + j;
#pragma unroll
        for (int r = 0; r < 8; ++r) {
            const int row = rbase + rt * 16 + r + 8 * hi;
            if (row < M) C[(size_t)row * NOUT + cb + l15] = acc[j][r];
        }
    }
}

// ---------------------------------------------------------------------------
// agg[i][f] = dinv[i]^2 * h[i][f]   (self-loop term; also initializes agg)
// ---------------------------------------------------------------------------
template <int F>
__global__ void k_selfinit(const float* __restrict__ h, const float* __restrict__ dinv,
                           float* __restrict__ agg, int N) {
    long long idx = (long long)blockIdx.x * blockDim.x + threadIdx.x;
    long long total = (long long)N * F;
    if (idx >= total) return;
    int node = (int)(idx / F);
    float d = dinv[node];
    agg[idx] = d * d * h[idx];
}

// ---------------------------------------------------------------------------
// Edge scatter: agg[col] += dinv[row]*dinv[col] * h[row], float4 granularity.
// Consecutive threads cover consecutive feature groups of one edge (coalesced).
// Atomics resolve in L2 (agg fits in the 192MB L2).
// ---------------------------------------------------------------------------
template <int F>
__global__ void k_scatter(const float* __restrict__ h, const long long* __restrict__ ei,
                          const float* __restrict__ dinv, float* __restrict__ agg,
                          long long E) {
    constexpr int G = F / 4;
    long long idx = (long long)blockIdx.x * blockDim.x + threadIdx.x;
    if (idx >= E * G) return;
    long long e = idx / G;
    int g = (int)(idx % G);
    long long r = ei[e];
    long long c = ei[E + e];
    float nrm = dinv[r] * dinv[c];
    const float4 v = *(const float4*)(h + r * F + g * 4);
    float* dst = agg + c * F + g * 4;
    atomicAdd(dst + 0, v.x * nrm);
    atomicAdd(dst + 1, v.y * nrm);
    atomicAdd(dst + 2, v.z * nrm);
    atomicAdd(dst + 3, v.w * nrm);
}

// ---------------------------------------------------------------------------
// out = (relu?) (agg + bias)
// ---------------------------------------------------------------------------
template <int F, bool RELU>
__global__ void k_finalize(const float* __restrict__ agg, const float* __restrict__ bias,
                           float* __restrict__ out, int N) {
    long long idx = (long long)blockIdx.x * blockDim.x + threadIdx.x;
    long long total = (long long)N * F;
    if (idx >= total) return;
    float v = agg[idx] + bias[idx & (F - 1)];
    if (RELU) v = v > 0.0f ? v : 0.0f;
    out[idx] = v;
}

// ---------------------------------------------------------------------------
// Fused link-prediction decoder. 16 label-edges per block (4 waves).
//   ef[e] = concat(z[src[e]], z[dst[e]])  (each endpoint = contiguous 128B,
//            gathered with async global->LDS b128 copies)
//   hidden = relu(ef @ lpW1 + lpb1)       (f32 WMMA, K=64)
//   out[e] = hidden . lpW2 + lpb2         (LDS ds_add_f32 reduction)
// ---------------------------------------------------------------------------
__global__ void __launch_bounds__(128)
k_decode(const float* __restrict__ z, const long long* __restrict__ eli,
         const float* __restrict__ lpW1, const float* __restrict__ lpb1,
         const float* __restrict__ lpW2, const float* __restrict__ lpb2,
         float* __restrict__ out, long long EL) {
    __shared__ __align__(16) float sW[64 * 64];   // 16 KB
    __shared__ __align__(16) float sEF[16 * 64];  // 4 KB
    __shared__ float sSum[16];
    const int tid  = threadIdx.x;
    const int lane = tid & 31;
    const int wave = tid >> 5;
    const long long ebase = (long long)blockIdx.x * 16;

    // stage lpW1 (contiguous 4096 floats) asynchronously
    {
        const unsigned lw = lds_addr(sW);
        for (int c = tid; c < 64 * 64 / 4; c += 128)
            async_b128(lw + c * 16, lpW1 + c * 4);
    }
    // gather edge features: 16 edges x 2 endpoints x 8 b128 chunks = 256
    {
        const unsigned le = lds_addr(sEF);
        for (int c = tid; c < 256; c += 128) {
            const int e  = c >> 4;        // edge within block
            const int f  = (c & 15) * 4;  // feature start (0..60)
            long long eg = ebase + e;
            if (eg >= EL) eg = EL - 1;    // clamp; rows beyond EL never stored
            const long long node = (f < 32) ? eli[eg] : eli[EL + eg];
            async_b128(le + c * 16, z + node * 32 + (f & 31));
        }
    }
    if (tid < 16) sSum[tid] = 0.0f;
    wait_async0();
    __syncthreads();

    const int hi  = lane >> 4;
    const int l15 = lane & 15;
    const int cb  = wave * 16;
    v8f acc = {};
#pragma unroll
    for (int k0 = 0; k0 < 64; k0 += 4) {
        const int ka = k0 + 2 * hi;
        v2f a, b;
        a.x = sEF[l15 * 64 + ka];
        a.y = sEF[l15 * 64 + ka + 1];
        b.x = sW[ka * 64 + cb + l15];
        b.y = sW[(ka + 1) * 64 + cb + l15];
        acc = __builtin_amdgcn_wmma_f32_16x16x4_f32(false, a, false, b,
                                                    (short)0, acc, false, false);
    }

    const float bn = lpb1[cb + l15];
    const float wn = lpW2[cb + l15];
#pragma unroll
    for (int r = 0; r < 8; ++r) {
        float hv = acc[r] + bn;
        hv = hv > 0.0f ? hv : 0.0f;
        atomicAdd(&sSum[r + 8 * hi], hv * wn);  // ds_add_f32
    }
    __syncthreads();

    if (tid < 16) {
        long long eg = ebase + tid;
        if (eg < EL) out[eg] = sSum[tid] + lpb2[0];
    }
}

// ---------------------------------------------------------------------------
// Host-side orchestration
// ---------------------------------------------------------------------------
extern "C" void kernel_launch(void* const* d_in, const int* in_sizes, int n_in,
                              void* d_out, int out_size, void* d_ws, size_t ws_size,
                              hipStream_t stream) {
    const float*     x    = (const float*)d_in[0];
    const long long* ei   = (const long long*)d_in[1];
    const long long* eli  = (const long long*)d_in[2];
    const float*     W1   = (const float*)d_in[3];
    const float*     b1   = (const float*)d_in[4];
    const float*     W2   = (const float*)d_in[5];
    const float*     b2   = (const float*)d_in[6];
    const float*     W3   = (const float*)d_in[7];
    const float*     b3   = (const float*)d_in[8];
    const float*     lpW1 = (const float*)d_in[9];
    const float*     lpb1 = (const float*)d_in[10];
    const float*     lpW2 = (const float*)d_in[11];
    const float*     lpb2 = (const float*)d_in[12];

    const int       N  = in_sizes[0] / 128;   // 100000
    const long long E  = in_sizes[1] / 2;     // 3.2M
    const long long EL = in_sizes[2] / 2;     // 1M

    float* out  = (float*)d_out;
    float* zout = out + EL;                   // z occupies d_out[EL .. EL + N*32)

    // workspace layout: dinv | bufA | bufB | bufC  (~77 MB)
    size_t off  = (((size_t)N * sizeof(float)) + 255) & ~(size_t)255;
    float* dinv = (float*)d_ws;
    float* bufA = (float*)((char*)d_ws + off);
    float* bufB = bufA + (size_t)N * 64;
    float* bufC = bufB + (size_t)N * 64;

    const int T = 256;
    int gN    = (N + T - 1) / T;
    int gE    = (int)((E + T - 1) / T);
    int gNF64 = (int)(((long long)N * 64 + T - 1) / T);
    int gNF32 = (int)(((long long)N * 32 + T - 1) / T);
    int gE16  = (int)((E * 16 + T - 1) / T);
    int gE8   = (int)((E * 8 + T - 1) / T);
    int gTile = (N + 63) / 64;                // 64 rows per GEMM block

    // GCN symmetric normalization
    k_fill1<<<gN, T, 0, stream>>>(dinv, N);
    k_degree<<<gE, T, 0, stream>>>(ei, dinv, E);
    k_rsqrt<<<gN, T, 0, stream>>>(dinv, N);

    // layer 1: 128 -> 64, relu
    k_gemm_wmma<128, 64><<<gTile, 128, 0, stream>>>(x, W1, bufA, N);
    k_selfinit<64><<<gNF64, T, 0, stream>>>(bufA, dinv, bufB, N);
    k_scatter<64><<<gE16, T, 0, stream>>>(bufA, ei, dinv, bufB, E);
    k_finalize<64, true><<<gNF64, T, 0, stream>>>(bufB, b1, bufC, N);

    // layer 2: 64 -> 64, relu
    k_gemm_wmma<64, 64><<<gTile, 128, 0, stream>>>(bufC, W2, bufA, N);
    k_selfinit<64><<<gNF64, T, 0, stream>>>(bufA, dinv, bufB, N);
    k_scatter<64><<<gE16, T, 0, stream>>>(bufA, ei, dinv, bufB, E);
    k_finalize<64, true><<<gNF64, T, 0, stream>>>(bufB, b2, bufC, N);

    // layer 3: 64 -> 32, no relu, directly into d_out z-region
    k_gemm_wmma<64, 32><<<gTile, 128, 0, stream>>>(bufC, W3, bufA, N);
    k_selfinit<32><<<gNF32, T, 0, stream>>>(bufA, dinv, bufB, N);
    k_scatter<32><<<gE8, T, 0, stream>>>(bufA, ei, dinv, bufB, E);
    k_finalize<32, false><<<gNF32, T, 0, stream>>>(bufB, b3, zout, N);

    // fused decoder -> link_pred in d_out[0..EL)
    int gDec = (int)((EL + 15) / 16);
    k_decode<<<gDec, 128, 0, stream>>>(zout, eli, lpW1, lpb1, lpW2, lpb2, out, EL);
}